// DifferentiableRicciLoss_73005854097883
// MI455X (gfx1250) — compile-verified
//
#include <hip/hip_runtime.h>

typedef __attribute__((ext_vector_type(2))) float v2f;
typedef __attribute__((ext_vector_type(8))) float v8f;

#define N_PTS 8192
#define DIMS  128
#define ROWS  64     // rows per block (output M tile)
#define COLS  128    // columns per iteration (output N tile)
#define LSTRIDE 132  // padded LDS row stride (floats), multiple of 4, bank-spread

// ---------------------------------------------------------------- squared norms
__global__ __launch_bounds__(256) void norms_kernel(const float* __restrict__ x0,
                                                    const float* __restrict__ x1,
                                                    float* __restrict__ sq0,
                                                    float* __restrict__ sq1) {
  const float* x = blockIdx.y ? x1 : x0;
  float* sq = blockIdx.y ? sq1 : sq0;
  int wave = threadIdx.x >> 5, lane = threadIdx.x & 31;
  int row = blockIdx.x * 8 + wave;                 // one wave per row
  float4 v = ((const float4*)(x + row * DIMS))[lane];
  float s = v.x * v.x + v.y * v.y + v.z * v.z + v.w * v.w;
  #pragma unroll
  for (int off = 16; off; off >>= 1) s += __shfl_xor(s, off, 32);
  if (lane == 0) sq[row] = s;
}

// ------------------------------------------- fused Gram (WMMA f32) + kNN select
__global__ __launch_bounds__(256) void knn_kernel(const float* __restrict__ x0,
                                                  const float* __restrict__ x1,
                                                  const float* __restrict__ sq0,
                                                  const float* __restrict__ sq1,
                                                  float* __restrict__ a0,
                                                  float* __restrict__ a1) {
  extern __shared__ float smem[];
  float* Atile = smem;                         // ROWS * LSTRIDE
  float* Btile = Atile + ROWS * LSTRIDE;       // COLS * LSTRIDE (reused as dist tile)
  float* sqB   = Btile + COLS * LSTRIDE;       // COLS
  float* sqA   = sqB + COLS;                   // ROWS
  float* knnb  = sqA + ROWS;                   // ROWS * 4 * 12 candidate lists

  const float* X  = blockIdx.y ? x1 : x0;
  const float* SQ = blockIdx.y ? sq1 : sq0;
  float* AOUT = blockIdx.y ? a1 : a0;

  const int tid = threadIdx.x;
  const int lane = tid & 31, wave = tid >> 5;
  const int rowBase = blockIdx.x * ROWS;

  // init candidate lists to +inf
  for (int i = tid; i < ROWS * 4 * 12; i += 256) knnb[i] = 3.0e38f;

  // stage A tile (64 x 128) + its squared norms
  for (int i = tid; i < ROWS * DIMS / 4; i += 256) {
    int r = i >> 5, c = (i & 31) << 2;
    *(float4*)(Atile + r * LSTRIDE + c) = *(const float4*)(X + (rowBase + r) * DIMS + c);
  }
  if (tid < ROWS) sqA[tid] = SQ[rowBase + tid];
  __syncthreads();

  const int selRow = tid >> 2, selSub = tid & 3;   // 4 selector threads per row
  const float mySqA = sqA[selRow];
  float myMax = 3.0e38f;

  const int mtile = wave >> 1;          // 4 M-tiles of 16
  const int ntb   = (wave & 1) * 4;     // 4 N-tiles of 16 per wave

  for (int ct = 0; ct < N_PTS / COLS; ++ct) {
    const int colBase = ct * COLS;
    __syncthreads();   // prior selection finished before Btile overwrite

    // stage B tile (128 x 128) + its squared norms
    for (int i = tid; i < COLS * DIMS / 4; i += 256) {
      int r = i >> 5, c = (i & 31) << 2;
      *(float4*)(Btile + r * LSTRIDE + c) = *(const float4*)(X + (colBase + r) * DIMS + c);
    }
    if (tid < COLS) sqB[tid] = SQ[colBase + tid];
    if (ct + 1 < N_PTS / COLS)   // warm L2/WGP$ for next tile
      __builtin_prefetch(X + (colBase + COLS) * DIMS, 0, 1);
    __syncthreads();

    // ---- WMMA: 4 output tiles of 16x16 per wave, K accumulated in 4-steps ----
    v8f acc0 = {}, acc1 = {}, acc2 = {}, acc3 = {};
    const int half = lane >> 4;                   // 0: K pair {0,1}, 1: {2,3}
    const float* Ap = Atile + (mtile * 16 + (lane & 15)) * LSTRIDE + 2 * half;
    const float* Bp = Btile + (ntb * 16 + (lane & 15)) * LSTRIDE + 2 * half;
    for (int k0 = 0; k0 < DIMS; k0 += 4) {
      v2f af = *(const v2f*)(Ap + k0);
      v2f b0 = *(const v2f*)(Bp + k0);
      v2f b1 = *(const v2f*)(Bp + 16 * LSTRIDE + k0);
      v2f b2 = *(const v2f*)(Bp + 32 * LSTRIDE + k0);
      v2f b3 = *(const v2f*)(Bp + 48 * LSTRIDE + k0);
      acc0 = __builtin_amdgcn_wmma_f32_16x16x4_f32(false, af, false, b0, (short)0, acc0, false, false);
      acc1 = __builtin_amdgcn_wmma_f32_16x16x4_f32(false, af, false, b1, (short)0, acc1, false, false);
      acc2 = __builtin_amdgcn_wmma_f32_16x16x4_f32(false, af, false, b2, (short)0, acc2, false, false);
      acc3 = __builtin_amdgcn_wmma_f32_16x16x4_f32(false, af, false, b3, (short)0, acc3, false, false);
    }
    __syncthreads();   // all waves done reading Btile

    // spill dot products into LDS dist staging (reusing Btile region)
    {
      const int rb = mtile * 16 + (half << 3);
      const int cb = (lane & 15);
      #pragma unroll
      for (int r = 0; r < 8; ++r) {
        Btile[(rb + r) * LSTRIDE + (ntb + 0) * 16 + cb] = acc0[r];
        Btile[(rb + r) * LSTRIDE + (ntb + 1) * 16 + cb] = acc1[r];
        Btile[(rb + r) * LSTRIDE + (ntb + 2) * 16 + cb] = acc2[r];
        Btile[(rb + r) * LSTRIDE + (ntb + 3) * 16 + cb] = acc3[r];
      }
    }
    __syncthreads();

    // ---- streaming top-11 select on squared distance (monotonic) ----
    {
      float* mylist = knnb + (selRow * 4 + selSub) * 12;
      const int cb = selSub * 32;
      for (int c = 0; c < 32; ++c) {
        float g = Btile[selRow * LSTRIDE + cb + c];
        float d2 = mySqA + sqB[cb + c] - 2.0f * g;
        if (d2 < myMax) {
          int im = 0; float vm = mylist[0];
          #pragma unroll
          for (int i = 1; i < 11; ++i) { float q = mylist[i]; if (q > vm) { vm = q; im = i; } }
          mylist[im] = d2;
          float nm = mylist[0];
          #pragma unroll
          for (int i = 1; i < 11; ++i) nm = fmaxf(nm, mylist[i]);
          myMax = nm;
        }
      }
    }
  }
  __syncthreads();

  // ---- merge 4 sub-lists -> 11 smallest sorted ascending; drop self; output a ----
  if (tid < ROWS) {
    float* lists = knnb + tid * 4 * 12;
    float sel[11];
    #pragma unroll
    for (int j = 0; j < 11; ++j) {
      float vm = 3.0e38f; int best = 0;
      for (int i = 0; i < 48; ++i) {
        if ((i % 12) < 11) { float q = lists[i]; if (q < vm) { vm = q; best = i; } }
      }
      lists[best] = 3.0e38f;
      sel[j] = vm;
    }
    float d[10]; float sum = 0.f;
    #pragma unroll
    for (int j = 0; j < 10; ++j) { d[j] = sqrtf(fmaxf(sel[j + 1], 1e-12f)); sum += d[j]; }
    float inv = 1.0f / (sum + 1e-10f);
    int grow = rowBase + tid;
    #pragma unroll
    for (int j = 0; j < 10; ++j) AOUT[grow * 10 + j] = d[j] * inv;
  }
}

// ---------------------------------------------------------------- Sinkhorn (1 lane / sample)
__global__ __launch_bounds__(256) void sinkhorn_kernel(const float* __restrict__ A,
                                                       const float* __restrict__ B,
                                                       float* __restrict__ partials) {
  const int idx = blockIdx.x * 256 + threadIdx.x;
  float a[10], b[10], u[10], v[10], kexp[10];
  #pragma unroll
  for (int i = 0; i < 10; ++i) {
    a[i] = A[idx * 10 + i];
    b[i] = B[idx * 10 + i];
    u[i] = 1.0f;
    kexp[i] = expf(-(float)i);    // K[i][j] = exp(-|i-j|/(k*reg)) = exp(-|i-j|)
  }
  for (int it = 0; it < 50; ++it) {
    #pragma unroll
    for (int j = 0; j < 10; ++j) {
      float s = 0.f;
      #pragma unroll
      for (int i = 0; i < 10; ++i) s += u[i] * kexp[i > j ? i - j : j - i];
      v[j] = b[j] / (s + 1e-10f);
    }
    #pragma unroll
    for (int i = 0; i < 10; ++i) {
      float s = 0.f;
      #pragma unroll
      for (int j = 0; j < 10; ++j) s += v[j] * kexp[i > j ? i - j : j - i];
      u[i] = a[i] / (s + 1e-10f);
    }
  }
  #pragma unroll
  for (int j = 0; j < 10; ++j) {
    float s = 0.f;
    #pragma unroll
    for (int i = 0; i < 10; ++i) s += u[i] * kexp[i > j ? i - j : j - i];
    v[j] = b[j] / (s + 1e-10f);
  }
  float per = 0.f;
  #pragma unroll
  for (int i = 0; i < 10; ++i)
    #pragma unroll
    for (int j = 0; j < 10; ++j) {
      int dd = i > j ? i - j : j - i;
      per += u[i] * kexp[dd] * (0.1f * (float)dd) * v[j];
    }

  __shared__ float red[256];
  red[threadIdx.x] = per;
  __syncthreads();
  #pragma unroll
  for (int s2 = 128; s2; s2 >>= 1) {
    if (threadIdx.x < s2) red[threadIdx.x] += red[threadIdx.x + s2];
    __syncthreads();
  }
  if (threadIdx.x == 0) partials[blockIdx.x] = red[0];
}

// ---------------------------------------------------------------- final deterministic mean
__global__ void final_reduce(const float* __restrict__ partials, float* __restrict__ out) {
  float s = partials[threadIdx.x];
  #pragma unroll
  for (int off = 16; off; off >>= 1) s += __shfl_xor(s, off, 32);
  if (threadIdx.x == 0) out[0] = s * (1.0f / 8192.0f);
}

extern "C" void kernel_launch(void* const* d_in, const int* in_sizes, int n_in,
                              void* d_out, int out_size, void* d_ws, size_t ws_size,
                              hipStream_t stream) {
  (void)in_sizes; (void)n_in; (void)out_size; (void)ws_size;
  const float* emb = (const float*)d_in[0];
  const float* ref = (const float*)d_in[1];
  float* ws   = (float*)d_ws;
  float* sq0  = ws;                       // 8192
  float* sq1  = ws + 8192;                // 8192
  float* aOut = ws + 16384;               // 8192*10
  float* bOut = aOut + 8192 * 10;         // 8192*10
  float* part = bOut + 8192 * 10;         // 32
  float* out  = (float*)d_out;

  norms_kernel<<<dim3(1024, 2), 256, 0, stream>>>(emb, ref, sq0, sq1);

  size_t lds = (size_t)(ROWS * LSTRIDE + COLS * LSTRIDE + COLS + ROWS + ROWS * 4 * 12) * sizeof(float);
  knn_kernel<<<dim3(N_PTS / ROWS, 2), 256, lds, stream>>>(emb, ref, sq0, sq1, aOut, bOut);

  sinkhorn_kernel<<<32, 256, 0, stream>>>(aOut, bOut, part);
  final_reduce<<<1, 32, 0, stream>>>(part, out);
}